// SoftIndex_80556406604040
// MI455X (gfx1250) — compile-verified
//
#include <hip/hip_runtime.h>
#include <hip/hip_bf16.h>

// ---------------------------------------------------------------------------
// Row-wise: out[i, :] = cumsum(tanh(diff(x[i, :]) with d[0]=0))
// Memory-bound streaming scan: 512 MB total traffic -> ~22 us @ 23.3 TB/s.
// One wave32 per row; 128 floats (4/lane) per iteration; 64 iterations/row.
// WMMA deliberately not used: the matmul formulation is 2*8192^3 ~ 1.1 TFLOP
// vs a 22 us memory floor for the O(N) scan -> ~500x slower. CDNA5-specific
// paths used instead: v_tanh_f32 (TRANS), wave32 ds_bpermute shuffle scan,
// v_readlane broadcasts (no DS / no wait on the serial carry chain),
// global_prefetch_b8, NT-hinted b128 loads/stores (L2 is 192 MB < 512 MB
// of single-use traffic).
// ---------------------------------------------------------------------------

// Native clang vector type: required by __builtin_nontemporal_{load,store}.
typedef float v4f __attribute__((ext_vector_type(4)));

#if __has_builtin(__builtin_amdgcn_tanhf)
__device__ __forceinline__ float fast_tanh(float a) {
    return __builtin_amdgcn_tanhf(a);           // gfx1250 v_tanh_f32 (TRANS op)
}
#elif __has_builtin(__builtin_amdgcn_tanh_f32)
__device__ __forceinline__ float fast_tanh(float a) {
    return __builtin_amdgcn_tanh_f32(a);
}
#else
__device__ __forceinline__ float fast_tanh(float a) {
    return tanhf(a);                            // ocml fallback
}
#endif

// Constant-lane broadcast via v_readlane_b32 (scalar result, no LDS traffic,
// no DScnt wait) instead of the ds_bpermute path that __shfl takes.
__device__ __forceinline__ float read_lane_f32(float v, int lane) {
    return __int_as_float(__builtin_amdgcn_readlane(__float_as_int(v), lane));
}

namespace {
constexpr int S               = 8192;   // row length
constexpr int B               = 8192;   // number of rows
constexpr int LANES           = 32;     // wave32
constexpr int VEC             = 4;      // floats per lane per iteration
constexpr int TILE            = LANES * VEC;   // 128 elements / wave / iter
constexpr int ITERS           = S / TILE;      // 64
constexpr int WAVES_PER_BLOCK = 8;             // 256 threads
}

__global__ __launch_bounds__(WAVES_PER_BLOCK * LANES)
void diff_tanh_cumsum_kernel(const float* __restrict__ x,
                             float* __restrict__ out) {
    const int wave = (int)(threadIdx.x >> 5);
    const int lane = (int)(threadIdx.x & 31u);
    const int row  = (int)blockIdx.x * WAVES_PER_BLOCK + wave;
    if (row >= B) return;

    const float* __restrict__ xrow = x   + (size_t)row * S;
    float*       __restrict__ orow = out + (size_t)row * S;

    // Running cumulative sum for this row, and the last raw x of the previous
    // tile (seeded with x[row][0] so that d[0] == 0 -> tanh(0) == 0).
    float carry     = 0.0f;
    float prev_last = xrow[0];

    for (int it = 0; it < ITERS; ++it) {
        const int base = it * TILE + lane * VEC;

        // Prefetch the next tile (gfx1250 global_prefetch_b8).
        if (it + 1 < ITERS) {
            __builtin_prefetch(xrow + base + TILE, 0, 0);
        }

        // Streamed 128-bit load with non-temporal hint.
        const v4f v =
            __builtin_nontemporal_load(reinterpret_cast<const v4f*>(xrow + base));

        // Left neighbor of this lane's first element: previous lane's last
        // element, or the carried value for lane 0.
        float from_left = __shfl_up(v.w, 1, LANES);
        if (lane == 0) from_left = prev_last;

        // Per-element tanh(diff).
        const float t0 = fast_tanh(v.x - from_left);
        const float t1 = fast_tanh(v.y - v.x);
        const float t2 = fast_tanh(v.z - v.y);
        const float t3 = fast_tanh(v.w - v.z);

        // In-register inclusive scan of the 4-element chunk.
        const float s0 = t0;
        const float s1 = s0 + t1;
        const float s2 = s1 + t2;
        const float s3 = s2 + t3;

        // Wave32 inclusive scan of per-lane chunk totals (5 shuffle steps).
        float inc = s3;
#pragma unroll
        for (int off = 1; off < LANES; off <<= 1) {
            const float n = __shfl_up(inc, off, LANES);
            if (lane >= off) inc += n;
        }

        // Exclusive offset for this lane within the tile + row carry.
        const float excl = carry + (inc - s3);

        v4f o;
        o.x = excl + s0;
        o.y = excl + s1;
        o.z = excl + s2;
        o.w = excl + s3;
        __builtin_nontemporal_store(o, reinterpret_cast<v4f*>(orow + base));

        // Propagate row state via v_readlane (scalar broadcast, no DS wait on
        // the serial carry chain).
        carry    += read_lane_f32(inc, LANES - 1);
        prev_last = read_lane_f32(v.w, LANES - 1);
    }
}

extern "C" void kernel_launch(void* const* d_in, const int* in_sizes, int n_in,
                              void* d_out, int out_size, void* d_ws, size_t ws_size,
                              hipStream_t stream) {
    (void)in_sizes; (void)n_in; (void)out_size; (void)d_ws; (void)ws_size;
    const float* x   = (const float*)d_in[0];
    float*       out = (float*)d_out;

    dim3 block(WAVES_PER_BLOCK * LANES);         // 256 threads = 8 wave32
    dim3 grid(B / WAVES_PER_BLOCK);              // 1024 blocks, 8192 waves
    diff_tanh_cumsum_kernel<<<grid, block, 0, stream>>>(x, out);
}